// Aggregator_49615462203694
// MI455X (gfx1250) — compile-verified
//
#include <hip/hip_runtime.h>

// N=16384, K=32, D=256; out[n,d] = curr_emb[n,0,d] + sum_k alpha[n,k]*msg[n,k,d]
// Memory-bound: ~546MB @ 23.3TB/s => ~24us floor. WMMA f32 16x16x4 carries the
// weighted reduction; NT loads keep the 512MB msg stream out of L2.

typedef __attribute__((ext_vector_type(2))) float v2f;
typedef __attribute__((ext_vector_type(8))) float v8f;

#define N_TOT 16384
#define KDIM 32
#define DDIM 256
#define WAVES_PER_BLOCK 8

__global__ __launch_bounds__(256) void agg_wmma_kernel(
    const float* __restrict__ curr_emb,   // [N, K, D]
    const float* __restrict__ alpha,      // [N, K] (trailing 1 dropped)
    const float* __restrict__ msg,        // [N, K, D]
    float* __restrict__ out)              // [N, D]
{
    const int lane = threadIdx.x & 31;
    const int wave = threadIdx.x >> 5;
    const int n    = blockIdx.x * WAVES_PER_BLOCK + wave;

    const size_t nKD  = (size_t)n * (KDIM * DDIM);
    const float* msg_n  = msg + nKD;
    const float* curr_n = curr_emb + nKD;          // k = 0 slice
    float*       out_n  = out + (size_t)n * DDIM;

    // One alpha value per lane (coalesced 128B per wave).
    const float alpha_l = alpha[(size_t)n * KDIM + lane];

    // A-matrix (16x4 f32) layout: VGPR0 holds K = kc + (lane<16 ? 0 : 2),
    // VGPR1 holds K = kc + (lane<16 ? 1 : 3); value independent of M row
    // => alpha automatically replicated across all 16 rows.
    const int hi = (lane & 16) ? 2 : 0;
    float a0[8], a1[8];
#pragma unroll
    for (int kc = 0; kc < 8; ++kc) {
        a0[kc] = __shfl(alpha_l, kc * 4 + hi,     32);
        a1[kc] = __shfl(alpha_l, kc * 4 + hi + 1, 32);
    }

    const int col   = lane & 15;              // column within a 16-wide d-tile
    const int krow0 = (lane & 16) ? 2 : 0;    // B VGPR0 K-row within 4-chunk

    // Process d in pairs of 16-wide tiles so the final store is a full
    // 32-lane coalesced 128B store.
#pragma unroll 2
    for (int dtp = 0; dtp < 8; ++dtp) {
        const int dA = dtp * 32 + col;
        const int dB = dA + 16;

        const float curA = curr_n[dA];
        const float curB = curr_n[dB];
        v8f c0, c1;
#pragma unroll
        for (int i = 0; i < 8; ++i) { c0[i] = curA; c1[i] = curB; }

#pragma unroll
        for (int kc = 0; kc < 8; ++kc) {
            const float* base = msg_n + (size_t)(kc * 4 + krow0) * DDIM;
            // B-matrix (4x16 f32): VGPR0 = row krow0, VGPR1 = row krow0+1;
            // 16 consecutive d per half-wave -> coalesced 64B segments.
            v2f bA, bB, aOp;
            bA.x = __builtin_nontemporal_load(base + dA);
            bA.y = __builtin_nontemporal_load(base + DDIM + dA);
            bB.x = __builtin_nontemporal_load(base + dB);
            bB.y = __builtin_nontemporal_load(base + DDIM + dB);
            aOp.x = a0[kc];
            aOp.y = a1[kc];
            // D = A x B + C : rows all identical = alpha . msg + curr
            c0 = __builtin_amdgcn_wmma_f32_16x16x4_f32(
                false, aOp, false, bA, (short)0, c0, false, false);
            c1 = __builtin_amdgcn_wmma_f32_16x16x4_f32(
                false, aOp, false, bB, (short)0, c1, false, false);
        }

        // D VGPR0: lanes 0-15 = (M=0, N=lane) of tile A -> d = dtp*32+lane;
        //          lanes 16-31 = (M=8, N=lane-16) of tile B -> d = dtp*32+lane.
        const float v = (lane < 16) ? c0[0] : c1[0];
        __builtin_nontemporal_store(v, out_n + dtp * 32 + lane);
    }
}

extern "C" void kernel_launch(void* const* d_in, const int* in_sizes, int n_in,
                              void* d_out, int out_size, void* d_ws, size_t ws_size,
                              hipStream_t stream) {
    const float* curr_emb = (const float*)d_in[0];
    const float* alpha    = (const float*)d_in[1];
    const float* msg      = (const float*)d_in[2];
    float*       out      = (float*)d_out;

    dim3 grid(N_TOT / WAVES_PER_BLOCK);   // 2048 blocks x 8 waves = 16384 n
    dim3 block(256);
    agg_wmma_kernel<<<grid, block, 0, stream>>>(curr_emb, alpha, msg, out);
}